// my_sageCN_4861902979404
// MI455X (gfx1250) — compile-verified
//
#include <hip/hip_runtime.h>
#include <hip/hip_bf16.h>
#include <math.h>

typedef __attribute__((ext_vector_type(2))) float v2f;
typedef __attribute__((ext_vector_type(8))) float v8f;

#define NN 50000
#define NE 1600000
#define NG 64

// ---------------------------------------------------------------------------
// CDNA5 async global->LDS copy (ASYNCcnt-tracked, bypasses VGPRs).
// VDST = 32-bit LDS byte offset (low 32 bits of generic shared pointer),
// VADDR = 64-bit global address, no SADDR ("off").
__device__ __forceinline__ void async_b128_to_lds(const void* lds_generic, const void* gptr) {
    unsigned int lds_off = (unsigned int)(unsigned long long)(uintptr_t)lds_generic;
    asm volatile("global_load_async_to_lds_b128 %0, %1, off"
                 :: "v"(lds_off), "v"((unsigned long long)(uintptr_t)gptr)
                 : "memory");
}
__device__ __forceinline__ void wait_asynccnt0() {
    asm volatile("s_wait_asynccnt 0x0" ::: "memory");
}

// ---------------------------------------------------------------------------
// degree: one atomic add per edge
__global__ void degree_kernel(const int* __restrict__ dst, float* __restrict__ deg, int nEdges) {
    int e = blockIdx.x * blockDim.x + threadIdx.x;
    if (e < nEdges) atomicAdd(&deg[dst[e]], 1.0f);
}

// scatter-add of source-node features onto dst rows (block per edge)
__global__ void scatter_add_kernel(const int* __restrict__ src, const int* __restrict__ dst,
                                   const float* __restrict__ X, float* __restrict__ agg, int F) {
    int e = blockIdx.x;
    int s = src[e], d = dst[e];
    const float* xs = X + (size_t)s * F;
    float*       ad = agg + (size_t)d * F;
    for (int f = threadIdx.x; f < F; f += blockDim.x)
        atomicAdd(&ad[f], xs[f]);
}

// agg -> mean (in place)
__global__ void mean_div_kernel(float* __restrict__ agg, const float* __restrict__ deg,
                                int F, long long total) {
    long long i = (long long)blockIdx.x * blockDim.x + threadIdx.x;
    if (i < total) {
        int node = (int)(i / F);
        agg[i] = agg[i] / fmaxf(deg[node], 1.0f);
    }
}

// ---------------------------------------------------------------------------
// fp32 WMMA GEMM:  C[m][o] (+=) sum_k A[m][k] * W[o][k]  (+ bias) (relu)
// Block = 256 threads = 8 waves; block tile 16 x 128, one 16x16 WMMA tile per
// wave. K chunked by 32; tiles double-buffered in LDS and filled with
// global_load_async_to_lds_b128 so chunk c+1's fetch overlaps chunk c's WMMAs.
// Requires K % 4 == 0 (true here: 900 / 1024 / 2048).
#define TM 16
#define TN 128
#define KC 32
#define KS 36   // padded row stride (floats): 144B, 16B-aligned, conflict-free

__global__ __launch_bounds__(256) void gemm_wmma_kernel(
        const float* __restrict__ A,      // [M,K]
        const float* __restrict__ W,      // [O,K]  (row-major weights; we use W^T)
        const float* __restrict__ bias,   // [O] or nullptr
        float* __restrict__ C,            // [M,O]
        int M, int K, int O, int accumulate, int doRelu)
{
    __shared__ __align__(16) float As[2][TM][KS];   // rows: K-contiguous
    __shared__ __align__(16) float Bs[2][TN][KS];

    const int tid  = threadIdx.x;
    const int wave = tid >> 5;
    const int lane = tid & 31;
    const int half = lane >> 4;           // 0: K pair {0,1}, 1: K pair {2,3}
    const int l16  = lane & 15;
    const int rowBase = blockIdx.y * TM;
    const int colBase = blockIdx.x * TN;
    const int n = (wave << 4) + l16;      // column within block tile

    const float4 z4 = {0.0f, 0.0f, 0.0f, 0.0f};

    // stage one K-chunk into buffer `buf` (async for valid 4-float slots,
    // plain zero-stores for the K tail)
    auto stageA = [&](int buf, int k0) {
        if (tid < TM * (KC / 4)) {                    // 128 slots, 1 per thread
            int r = tid >> 3, kq = tid & 7;
            int kg = k0 + 4 * kq;
            float* dp = &As[buf][r][4 * kq];
            if (kg + 4 <= K)
                async_b128_to_lds(dp, &A[(size_t)(rowBase + r) * K + kg]);
            else
                *(float4*)dp = z4;
        }
    };
    auto stageB = [&](int buf, int k0) {
        for (int s = tid; s < TN * (KC / 4); s += 256) {  // 1024 slots, 4 per thread
            int nn = s >> 3, kq = s & 7;
            int kg = k0 + 4 * kq;
            float* dp = &Bs[buf][nn][4 * kq];
            if (kg + 4 <= K)
                async_b128_to_lds(dp, &W[(size_t)(colBase + nn) * K + kg]);
            else
                *(float4*)dp = z4;
        }
    };

    v8f c = {};

    const int nChunks = (K + KC - 1) / KC;
    stageA(0, 0);
    stageB(0, 0);

    for (int ch = 0; ch < nChunks; ++ch) {
        const int p = ch & 1;
        wait_asynccnt0();          // this wave's async fills are complete
        __syncthreads();           // -> every wave's fills of buf p are complete

        if (ch + 1 < nChunks) {    // prefetch next chunk under this chunk's math
            stageA(p ^ 1, (ch + 1) * KC);
            stageB(p ^ 1, (ch + 1) * KC);
        }

#pragma unroll
        for (int kk = 0; kk < KC; kk += 4) {
            v2f a, b;
            // A 16x4 frag: lanes 0-15 hold K={kk,kk+1}, lanes 16-31 K={kk+2,kk+3}
            const float* ap = &As[p][l16][kk + 2 * half];
            a[0] = ap[0]; a[1] = ap[1];
            // B 4x16 frag mirrors C/D row split: VGPR0 rows {0,2}, VGPR1 rows {1,3}
            const float* bp = &Bs[p][n][kk + 2 * half];
            b[0] = bp[0]; b[1] = bp[1];
            c = __builtin_amdgcn_wmma_f32_16x16x4_f32(
                    /*neg_a=*/false, a, /*neg_b=*/false, b,
                    /*c_mod=*/(short)0, c, /*reuse_a=*/false, /*reuse_b=*/false);
        }
    }

    // C/D layout: VGPR r -> row r (lanes 0-15) / row r+8 (lanes 16-31)
    const int col   = colBase + n;
    const int rBase = rowBase + (half ? 8 : 0);
#pragma unroll
    for (int r = 0; r < 8; ++r) {
        size_t idx = (size_t)(rBase + r) * O + col;
        float v = c[r];
        if (accumulate) v += C[idx];
        if (bias)       v += bias[col];
        if (doRelu)     v = fmaxf(v, 0.0f);
        C[idx] = v;
    }
}

// ---------------------------------------------------------------------------
// segment max over sorted batch ids; empty segment -> 0 (start < end guard)
__global__ void seg_max_kernel(const float* __restrict__ H, const int* __restrict__ batch,
                               float* __restrict__ pooled, int F, int nNodes) {
    int g = blockIdx.y;
    int f = blockIdx.x * blockDim.x + threadIdx.x;
    int lo = 0, hi = nNodes;
    while (lo < hi) { int mid = (lo + hi) >> 1; if (batch[mid] < g) lo = mid + 1; else hi = mid; }
    int start = lo;
    hi = nNodes;
    while (lo < hi) { int mid = (lo + hi) >> 1; if (batch[mid] < g + 1) lo = mid + 1; else hi = mid; }
    int end = lo;

    float m = 0.0f;
    if (start < end) {
        m = -INFINITY;
        for (int node = start; node < end; ++node)
            m = fmaxf(m, H[(size_t)node * F + f]);
    }
    pooled[(size_t)g * F + f] = m;
}

// tiny final projection: [64,1024] @ [1024,6]^T + b
__global__ void head_out_kernel(const float* __restrict__ t1, const float* __restrict__ Wf2,
                                const float* __restrict__ bf2, float* __restrict__ out) {
    int g = blockIdx.x;
    int o = threadIdx.x;
    if (o < 6) {
        float acc = bf2[o];
        const float* tr = t1 + (size_t)g * 1024;
        const float* wr = Wf2 + (size_t)o * 1024;
        for (int k = 0; k < 1024; ++k) acc += tr[k] * wr[k];
        out[g * 6 + o] = acc;
    }
}

// ---------------------------------------------------------------------------
extern "C" void kernel_launch(void* const* d_in, const int* in_sizes, int n_in,
                              void* d_out, int out_size, void* d_ws, size_t ws_size,
                              hipStream_t stream) {
    (void)in_sizes; (void)n_in; (void)out_size; (void)ws_size;

    const float* x    = (const float*)d_in[0];
    const int*   ei   = (const int*)d_in[1];
    const int*   bat  = (const int*)d_in[2];
    const float* W1l  = (const float*)d_in[3];
    const float* b1   = (const float*)d_in[4];
    const float* W1r  = (const float*)d_in[5];
    const float* W2l  = (const float*)d_in[6];
    const float* b2   = (const float*)d_in[7];
    const float* W2r  = (const float*)d_in[8];
    const float* Wf1  = (const float*)d_in[9];
    const float* bf1  = (const float*)d_in[10];
    const float* Wf2  = (const float*)d_in[11];
    const float* bf2  = (const float*)d_in[12];
    float* out = (float*)d_out;

    const int* src = ei;            // edge_index row 0
    const int* dst = ei + NE;       // edge_index row 1

    // workspace layout
    char* ws = (char*)d_ws;
    size_t off = 0;
    float* deg    = (float*)(ws + off); off += ((size_t)NN * 4 + 255) & ~(size_t)255;
    float* agg1   = (float*)(ws + off); off += (size_t)NN * 900  * 4;   // -> mean1
    float* h1     = (float*)(ws + off); off += (size_t)NN * 1024 * 4;
    float* agg2   = (float*)(ws + off); off += (size_t)NN * 1024 * 4;   // -> mean2
    float* h2     = (float*)(ws + off); off += (size_t)NN * 2048 * 4;
    float* pooled = (float*)(ws + off); off += (size_t)NG * 2048 * 4;
    float* t1     = (float*)(ws + off); off += (size_t)NG * 1024 * 4;

    // zero accumulators (async memset is graph-capture safe)
    hipMemsetAsync(deg,  0, (size_t)NN * 4,         stream);
    hipMemsetAsync(agg1, 0, (size_t)NN * 900 * 4,   stream);
    hipMemsetAsync(agg2, 0, (size_t)NN * 1024 * 4,  stream);

    // ---- layer 1: mean aggregation ----
    degree_kernel<<<(NE + 255) / 256, 256, 0, stream>>>(dst, deg, NE);
    scatter_add_kernel<<<NE, 256, 0, stream>>>(src, dst, x, agg1, 900);
    {
        long long total = (long long)NN * 900;
        mean_div_kernel<<<(unsigned)((total + 255) / 256), 256, 0, stream>>>(agg1, deg, 900, total);
    }
    // h1 = relu(mean1 @ W1l.T + b1 + x @ W1r.T)
    gemm_wmma_kernel<<<dim3(1024 / TN, NN / TM), 256, 0, stream>>>(agg1, W1l, b1,      h1, NN, 900, 1024, 0, 0);
    gemm_wmma_kernel<<<dim3(1024 / TN, NN / TM), 256, 0, stream>>>(x,    W1r, nullptr, h1, NN, 900, 1024, 1, 1);

    // ---- layer 2: mean aggregation of h1 ----
    scatter_add_kernel<<<NE, 256, 0, stream>>>(src, dst, h1, agg2, 1024);
    {
        long long total = (long long)NN * 1024;
        mean_div_kernel<<<(unsigned)((total + 255) / 256), 256, 0, stream>>>(agg2, deg, 1024, total);
    }
    // h2 = mean2 @ W2l.T + b2 + h1 @ W2r.T
    gemm_wmma_kernel<<<dim3(2048 / TN, NN / TM), 256, 0, stream>>>(agg2, W2l, b2,      h2, NN, 1024, 2048, 0, 0);
    gemm_wmma_kernel<<<dim3(2048 / TN, NN / TM), 256, 0, stream>>>(h1,   W2r, nullptr, h2, NN, 1024, 2048, 1, 0);

    // ---- global max pool ----
    seg_max_kernel<<<dim3(2048 / 256, NG), 256, 0, stream>>>(h2, bat, pooled, 2048, NN);

    // ---- MLP head ----
    gemm_wmma_kernel<<<dim3(1024 / TN, NG / TM), 256, 0, stream>>>(pooled, Wf1, bf1, t1, NG, 2048, 1024, 0, 1);
    head_out_kernel<<<NG, 32, 0, stream>>>(t1, Wf2, bf2, out);
}